// LIF_39144331936016
// MI455X (gfx1250) — compile-verified
//
#include <hip/hip_runtime.h>
#include <cstdint>
#include <cstddef>

// LIF scan over T for x_seq[B,T,C,H,W] (f32): u = 0.5u + x; o = (u>1); u -= o.
// Memory-bound streaming kernel (~268 MB @ 23.3 TB/s ~= 11.5 us floor).
// Uses CDNA5 async global->LDS DMA (ASYNCcnt) as a 4-deep per-wave prefetch
// pipeline; per-lane LDS slots, no barriers needed.

typedef float f32x4 __attribute__((ext_vector_type(4)));

#define LAM_F    0.5f
#define THRESH_F 1.0f

constexpr int Bn = 16, Tt = 16, Cc = 128, Hh = 32, Ww = 32;
constexpr int CHW     = Cc * Hh * Ww;          // 131072 elements (= 512 KiB time stride)
constexpr int TCHW    = Tt * CHW;              // 2097152
constexpr int VEC     = 4;
constexpr int GROUPS  = Bn * (CHW / VEC);      // 524288 float4 groups
constexpr int THREADS = 256;                   // 8 wave32s per block
constexpr int DEPTH   = 4;                     // async pipeline depth (power of 2)

__device__ __forceinline__ void async_load_b128(uint32_t lds_addr, const void* gaddr) {
    // CDNA5 async DMA: 16 bytes per lane, global -> LDS, tracked by ASYNCcnt.
    asm volatile("global_load_async_to_lds_b128 %0, %1, off"
                 :
                 : "v"(lds_addr), "v"(gaddr)
                 : "memory");
}

__global__ __launch_bounds__(THREADS) void lif_async_kernel(const float* __restrict__ x,
                                                            float* __restrict__ o) {
    __shared__ f32x4 stage[DEPTH][THREADS];    // 16 KiB / block, per-lane slots

    const int tid = threadIdx.x;
    const int g   = blockIdx.x * THREADS + tid;     // float4 group id (< GROUPS exactly)
    const int b   = g >> 15;                        // g / (CHW/VEC) ; CHW/VEC = 32768
    const int r   = g & 32767;                      // float4 index inside [C,H,W]
    const size_t base = (size_t)b * TCHW + (size_t)r * VEC;

    const float* xb = x + base;
    float*       ob = o + base;

    uint32_t slot[DEPTH];
#pragma unroll
    for (int s = 0; s < DEPTH; ++s)
        slot[s] = (uint32_t)(size_t)&stage[s][tid];   // low 32 bits = LDS offset

    // Prologue: prefetch time steps 0..DEPTH-1 (ASYNCcnt = DEPTH outstanding).
#pragma unroll
    for (int s = 0; s < DEPTH; ++s)
        async_load_b128(slot[s], (const void*)(xb + (size_t)s * CHW));

    f32x4 u = (f32x4){0.f, 0.f, 0.f, 0.f};

#pragma unroll
    for (int t = 0; t < Tt; ++t) {
        // Outstanding async loads at this point = min(DEPTH, T - t). Loads retire
        // in order, so waiting ASYNCcnt <= outstanding-1 guarantees tile t landed.
        const int rem = (Tt - t) < DEPTH ? (Tt - t) : DEPTH;   // constant after unroll
        if      (rem >= 4) asm volatile("s_wait_asynccnt 0x3" ::: "memory");
        else if (rem == 3) asm volatile("s_wait_asynccnt 0x2" ::: "memory");
        else if (rem == 2) asm volatile("s_wait_asynccnt 0x1" ::: "memory");
        else               asm volatile("s_wait_asynccnt 0x0" ::: "memory");

        f32x4 xt = stage[t & (DEPTH - 1)][tid];    // ds_load_b128

        // LIF update (4 neurons per thread)
        u = LAM_F * u + xt;
        f32x4 spike;
        spike.x = (u.x > THRESH_F) ? 1.f : 0.f;
        spike.y = (u.y > THRESH_F) ? 1.f : 0.f;
        spike.z = (u.z > THRESH_F) ? 1.f : 0.f;
        spike.w = (u.w > THRESH_F) ? 1.f : 0.f;
        u -= spike;

        // Streaming store: output never re-read; 256 MB working set > 192 MB L2.
        __builtin_nontemporal_store(spike, (f32x4*)(ob + (size_t)t * CHW));

        // Refill the slot we just consumed with time step t+DEPTH.
        if (t + DEPTH < Tt) {
            // Order the ds_load read above against the async LDS overwrite.
            asm volatile("s_wait_dscnt 0x0" ::: "memory");
            async_load_b128(slot[t & (DEPTH - 1)],
                            (const void*)(xb + (size_t)(t + DEPTH) * CHW));
        }
    }
}

extern "C" void kernel_launch(void* const* d_in, const int* in_sizes, int n_in,
                              void* d_out, int out_size, void* d_ws, size_t ws_size,
                              hipStream_t stream) {
    (void)in_sizes; (void)n_in; (void)d_ws; (void)ws_size; (void)out_size;
    const float* x = (const float*)d_in[0];   // x_seq [16,16,128,32,32] f32
    // d_in[1] (noise) is unused in the forward reference.
    float* o = (float*)d_out;                 // o [16,16,128,32,32] f32

    const int blocks = GROUPS / THREADS;      // 524288 / 256 = 2048
    lif_async_kernel<<<dim3(blocks), dim3(THREADS), 0, stream>>>(x, o);
}